// GraphStructureAttention_81853486727379
// MI455X (gfx1250) — compile-verified
//
#include <hip/hip_runtime.h>
#include <math.h>

#define NNODES 50000
#define NEDGES 800000
#define DMODEL 128
#define NHEADS 8
#define HEADDIM 16
#define SCALEF 0.25f /* 1/sqrt(16) */

typedef __attribute__((ext_vector_type(2))) float v2f;
typedef __attribute__((ext_vector_type(8))) float v8f;

// ---------- ordered-uint encoding for float atomic max ----------
__device__ __forceinline__ unsigned f2ord(float f) {
    unsigned u = __float_as_uint(f);
    return ((int)u >= 0) ? (u | 0x80000000u) : ~u;
}
__device__ __forceinline__ float ord2f(unsigned u) {
    unsigned v = (u & 0x80000000u) ? (u & 0x7FFFFFFFu) : ~u;
    return __uint_as_float(v);
}
#define ORD_NEG_INF 0x007FFFFFu  /* f2ord(-inf) */

// ---------- fp32 WMMA GEMM: C[M x Nout] = A[M x 128] * B[128 x Nout] + bias ----------
// grid.x = M/16 (M must be multiple of 16), grid.y = ceil(Nout/16), block = 32 (one wave)
__global__ __launch_bounds__(32)
void wmma_gemm_bias(const float* __restrict__ A, const float* __restrict__ B,
                    const float* __restrict__ bias, float* __restrict__ C,
                    int Nout, int ldc) {
    const int lane = threadIdx.x;
    const int half = lane >> 4;      // 0: K+0/K+1   1: K+2/K+3
    const int l    = lane & 15;
    const int m0   = blockIdx.x << 4;
    const int n0   = blockIdx.y << 4;

    const int  bcol_raw = n0 + l;
    const bool inb      = bcol_raw < Nout;
    const int  bcol     = inb ? bcol_raw : 0;
    const float bmask   = inb ? 1.0f : 0.0f;

    const float* aptr = A + (size_t)(m0 + l) * DMODEL + half * 2;
    const float* bptr = B + (size_t)(half * 2) * Nout + bcol;

    v8f acc = {};
#pragma unroll 8
    for (int k = 0; k < DMODEL; k += 4) {
        v2f a;
        a.x = aptr[k + 0];
        a.y = aptr[k + 1];
        v2f b;
        b.x = bptr[(size_t)k * Nout] * bmask;
        b.y = bptr[(size_t)(k + 1) * Nout] * bmask;
        // D = A(16x4,f32) * B(4x16,f32) + C(16x16,f32)
        acc = __builtin_amdgcn_wmma_f32_16x16x4_f32(
            /*neg_a=*/false, a, /*neg_b=*/false, b,
            /*c_mod=*/(short)0, acc, /*reuse_a=*/false, /*reuse_b=*/false);
    }

    if (inb) {
        const float bv = bias[bcol_raw];
#pragma unroll
        for (int r = 0; r < 8; ++r) {
            const int rowi = m0 + r + half * 8;   // VGPR r holds rows r and r+8
            C[(size_t)rowi * ldc + bcol_raw] = acc[r] + bv;
        }
    }
}

// ---------- init segment buffers ----------
__global__ __launch_bounds__(256)
void init_kernel(unsigned* __restrict__ segmax_u, float* __restrict__ denom,
                 float* __restrict__ agg) {
    const int i = blockIdx.x * blockDim.x + threadIdx.x;
    if (i < NNODES * NHEADS) { segmax_u[i] = ORD_NEG_INF; denom[i] = 0.0f; }
    if (i < NNODES * DMODEL) { agg[i] = 0.0f; }
}

// ---------- per-(edge,head) attention score + segment max ----------
__global__ __launch_bounds__(256)
void edge_score_kernel(const int* __restrict__ row, const int* __restrict__ col,
                       const float* __restrict__ Q, const float* __restrict__ Km,
                       float* __restrict__ sc, unsigned* __restrict__ segmax_u) {
    const int idx = blockIdx.x * blockDim.x + threadIdx.x;
    if (idx >= NEDGES * NHEADS) return;
    const int e = idx >> 3, h = idx & 7;
    const int r = row[e], c = col[e];
    const float4* q = (const float4*)(Q  + (size_t)r * DMODEL + h * HEADDIM);
    const float4* k = (const float4*)(Km + (size_t)c * DMODEL + h * HEADDIM);
    float s = 0.0f;
#pragma unroll
    for (int i = 0; i < 4; ++i) {
        const float4 qa = q[i], ka = k[i];
        s += qa.x * ka.x + qa.y * ka.y + qa.z * ka.z + qa.w * ka.w;
    }
    s = s * SCALEF + sc[idx];          // sc currently holds edge bias (E@We+be)
    sc[idx] = s;
    atomicMax(&segmax_u[r * NHEADS + h], f2ord(s));
}

// ---------- exp(score - segmax) + segment sum ----------
__global__ __launch_bounds__(256)
void edge_exp_kernel(const int* __restrict__ row, float* __restrict__ sc,
                     const unsigned* __restrict__ segmax_u, float* __restrict__ denom) {
    const int idx = blockIdx.x * blockDim.x + threadIdx.x;
    if (idx >= NEDGES * NHEADS) return;
    const int e = idx >> 3, h = idx & 7;
    const int r = row[e];
    const float m  = ord2f(segmax_u[r * NHEADS + h]);
    const float ex = expf(sc[idx] - m);
    sc[idx] = ex;
    atomicAdd(&denom[r * NHEADS + h], ex);
}

// ---------- weighted aggregation: agg[row] += att * V[col] ----------
// one thread per (edge, 4 consecutive feature dims)
__global__ __launch_bounds__(256)
void aggregate_kernel(const int* __restrict__ row, const int* __restrict__ col,
                      const float* __restrict__ sc, const float* __restrict__ denom,
                      const float* __restrict__ V, float* __restrict__ agg) {
    const int idx = blockIdx.x * blockDim.x + threadIdx.x;
    if (idx >= NEDGES * 32) return;
    const int e  = idx >> 5;
    const int t  = idx & 31;
    const int d0 = t * 4;
    const int h  = d0 >> 4;
    const int r = row[e], c = col[e];
    const float att = sc[e * NHEADS + h] / denom[r * NHEADS + h];
    const float4 v = *(const float4*)(V + (size_t)c * DMODEL + d0);
    float* dst = agg + (size_t)r * DMODEL + d0;
    atomicAdd(dst + 0, att * v.x);
    atomicAdd(dst + 1, att * v.y);
    atomicAdd(dst + 2, att * v.z);
    atomicAdd(dst + 3, att * v.w);
}

extern "C" void kernel_launch(void* const* d_in, const int* in_sizes, int n_in,
                              void* d_out, int out_size, void* d_ws, size_t ws_size,
                              hipStream_t stream) {
    const float* node  = (const float*)d_in[0];
    const int*   eidx  = (const int*)  d_in[1];
    const float* efeat = (const float*)d_in[2];
    const float* Wq = (const float*)d_in[3];
    const float* bq = (const float*)d_in[4];
    const float* Wk = (const float*)d_in[5];
    const float* bk = (const float*)d_in[6];
    const float* Wv = (const float*)d_in[7];
    const float* bv = (const float*)d_in[8];
    const float* We = (const float*)d_in[9];
    const float* be = (const float*)d_in[10];
    const float* Wo = (const float*)d_in[11];
    const float* bo = (const float*)d_in[12];
    float* out = (float*)d_out;

    const int* row = eidx;
    const int* col = eidx + NEDGES;

    // workspace layout (fp32): Q | K | V | sc | segmax(u32) | denom | agg
    float*    Qb     = (float*)d_ws;
    float*    Kb     = Qb + (size_t)NNODES * DMODEL;
    float*    Vb     = Kb + (size_t)NNODES * DMODEL;
    float*    sc     = Vb + (size_t)NNODES * DMODEL;
    unsigned* segmax = (unsigned*)(sc + (size_t)NEDGES * NHEADS);
    float*    denom  = (float*)(segmax + (size_t)NNODES * NHEADS);
    float*    agg    = denom + (size_t)NNODES * NHEADS;

    const dim3 wblk(32);
    const dim3 gN(NNODES / 16, DMODEL / 16);   // 3125 x 8 tiles
    const dim3 gE(NEDGES / 16, 1);             // 50000 x 1 tiles (Nout=8 masked)

    // init segment buffers (ws is poisoned; must re-init every call)
    {
        const int n = NNODES * DMODEL;
        init_kernel<<<(n + 255) / 256, 256, 0, stream>>>(segmax, denom, agg);
    }

    // dense projections via fp32 WMMA
    wmma_gemm_bias<<<gN, wblk, 0, stream>>>(node,  Wq, bq, Qb, DMODEL, DMODEL);
    wmma_gemm_bias<<<gN, wblk, 0, stream>>>(node,  Wk, bk, Kb, DMODEL, DMODEL);
    wmma_gemm_bias<<<gN, wblk, 0, stream>>>(node,  Wv, bv, Vb, DMODEL, DMODEL);
    wmma_gemm_bias<<<gE, wblk, 0, stream>>>(efeat, We, be, sc, NHEADS, NHEADS);

    // edge scores + segment softmax + aggregation
    {
        const int n = NEDGES * NHEADS;
        edge_score_kernel<<<(n + 255) / 256, 256, 0, stream>>>(row, col, Qb, Kb, sc, segmax);
        edge_exp_kernel<<<(n + 255) / 256, 256, 0, stream>>>(row, sc, segmax, denom);
    }
    {
        const int n = NEDGES * 32;
        aggregate_kernel<<<(n + 255) / 256, 256, 0, stream>>>(row, col, sc, denom, Vb, agg);
    }

    // output projection
    wmma_gemm_bias<<<gN, wblk, 0, stream>>>(agg, Wo, bo, out, DMODEL, DMODEL);
}